// Group_88321707475105
// MI455X (gfx1250) — compile-verified
//
#include <hip/hip_runtime.h>
#include <hip/hip_bf16.h>

// Problem constants (match the JAX reference)
static constexpr int kB = 32;    // batches
static constexpr int kN = 8192;  // points per batch
static constexpr int kG = 512;   // FPS centers (NUM_GROUP)
static constexpr int kK = 32;    // neighbors (GROUP_SIZE)

typedef __attribute__((ext_vector_type(2))) float v2f;
typedef __attribute__((ext_vector_type(8))) float v8f;

// ---------------------------------------------------------------------------
// Kernel 1: farthest point sampling. One block per batch, 1024 threads.
// Each thread owns 8 points in registers. Per iteration: uniform centroid
// load, register dist update, wave32 shuffle argmax, 32-wave LDS combine.
// Also emits a packed {x,y,z,|x|^2} float4 per point into d_ws (single
// coalesced b128 store) and the center coords into d_out's center region.
// ---------------------------------------------------------------------------
extern "C" __global__ __launch_bounds__(1024)
void fps_kernel(const float* __restrict__ xyz,
                float4* __restrict__ x4ws,      // [kB * kN] {x,y,z,|x|^2}
                float* __restrict__ centers) {  // d_out + kB*kG*kK*3
  const int b   = blockIdx.x;
  const int tid = threadIdx.x;                  // 0..1023
  constexpr int PT = kN / 1024;                 // 8 points per thread
  const float* xb = xyz + (size_t)b * kN * 3;

  float px[PT], py[PT], pz[PT], md[PT];
#pragma unroll
  for (int i = 0; i < PT; ++i) {
    int n = tid + i * 1024;                     // coalesced stride
    px[i] = xb[n * 3 + 0];
    py[i] = xb[n * 3 + 1];
    pz[i] = xb[n * 3 + 2];
    md[i] = 1e10f;
    float x2 = px[i] * px[i] + py[i] * py[i] + pz[i] * pz[i];
    x4ws[(size_t)b * kN + n] = make_float4(px[i], py[i], pz[i], x2);
  }

  __shared__ float s_v[32];
  __shared__ int   s_i[32];
  __shared__ int   s_best;

  int f = 0;                                    // reference starts at index 0
  for (int it = 0; it < kG; ++it) {
    float cx = xb[f * 3 + 0], cy = xb[f * 3 + 1], cz = xb[f * 3 + 2];
    if (tid == 0) {
      size_t o = ((size_t)b * kG + it) * 3;
      centers[o + 0] = cx; centers[o + 1] = cy; centers[o + 2] = cz;
    }

    float bestv = -1.0f; int besti = 0;
#pragma unroll
    for (int i = 0; i < PT; ++i) {
      float dx = px[i] - cx, dy = py[i] - cy, dz = pz[i] - cz;
      float d = dx * dx + dy * dy + dz * dz;
      md[i] = fminf(md[i], d);
      if (md[i] > bestv) { bestv = md[i]; besti = tid + i * 1024; }
    }

#pragma unroll
    for (int off = 16; off > 0; off >>= 1) {
      float ov = __shfl_down(bestv, off);
      int   oi = __shfl_down(besti, off);
      if (ov > bestv || (ov == bestv && oi < besti)) { bestv = ov; besti = oi; }
    }
    if ((tid & 31) == 0) { s_v[tid >> 5] = bestv; s_i[tid >> 5] = besti; }
    __syncthreads();
    if (tid < 32) {
      bestv = s_v[tid]; besti = s_i[tid];
#pragma unroll
      for (int off = 16; off > 0; off >>= 1) {
        float ov = __shfl_down(bestv, off);
        int   oi = __shfl_down(besti, off);
        if (ov > bestv || (ov == bestv && oi < besti)) { bestv = ov; besti = oi; }
      }
      if (tid == 0) s_best = besti;
    }
    __syncthreads();
    f = s_best;
  }
}

// Rare-path insert of candidate bits MASK into row R's kept set.
// Column j's score lives in (j<16 ? C0V : C1V) at lane (j&15)+((R&8)?16:0).
#define INSERT_ROW(R, MASK, C0V, C1V)                                        \
  {                                                                          \
    unsigned m_ = (MASK);                                                    \
    while (m_) {                                                             \
      int j_ = __ffs(m_) - 1; m_ &= m_ - 1;                                  \
      int sl_ = (j_ & 15) + (((R) & 8) ? 16 : 0);                            \
      float src_ = (j_ < 16) ? (C0V) : (C1V);                                \
      float cv_ = __shfl(src_, sl_);                                         \
      if (cv_ < rmax[(R)]) { /* rmax only decreases; recheck is exact */     \
        float mv_ = keptv[(R)]; int ml_ = lane;                              \
        _Pragma("unroll")                                                    \
        for (int off_ = 16; off_ > 0; off_ >>= 1) {                          \
          float ov_ = __shfl_down(mv_, off_);                                \
          int   ol_ = __shfl_down(ml_, off_);                                \
          if (ov_ > mv_) { mv_ = ov_; ml_ = ol_; }                           \
        }                                                                    \
        ml_ = __shfl(ml_, 0);                                                \
        if (lane == ml_) { keptv[(R)] = cv_; kepti[(R)] = n0 + j_; }         \
        float t_ = keptv[(R)];                                               \
        _Pragma("unroll")                                                    \
        for (int off_ = 16; off_ > 0; off_ >>= 1)                            \
          t_ = fmaxf(t_, __shfl_down(t_, off_));                             \
        rmax[(R)] = __shfl(t_, 0);                                           \
      }                                                                      \
    }                                                                        \
  }

// ---------------------------------------------------------------------------
// Kernel 2: KNN + grouping. One wave per (batch, 16-center tile).
// Score s(m,n) = -2 c_m . x_n + |x_n|^2 (rank-equal to sq-dist) via
// V_WMMA_F32_16X16X4_F32: A[16x4]=[-2cx,-2cy,-2cz,1], B[4x16]=[x;y;z;|x|^2].
// Candidate screening ballots directly on the native WMMA C/D layout
// (VGPR i = rows i / i+8 across the two half-waves) -> zero LDS traffic in
// the steady-state skip path. Exact top-32 kept one-per-lane; final bitonic
// sort reproduces top_k's stable (value, index) order before the gather.
// ---------------------------------------------------------------------------
extern "C" __global__ __launch_bounds__(32)
void knn_group_kernel(const float4* __restrict__ x4,
                      const float* __restrict__ centers,
                      float* __restrict__ nbr) {
  const int b    = blockIdx.x >> 5;             // 32 tiles per batch
  const int gt   = blockIdx.x & 31;
  const int lane = threadIdx.x;                 // 0..31 (wave32)
  const int g0   = gt * 16;
  const float4* p4 = x4 + (size_t)b * kN;
  const float*  cb = centers + ((size_t)b * kG + g0) * 3;

  // A matrix (16x4 f32): v[0] = K0 (lanes 0-15) / K2 (lanes 16-31),
  //                      v[1] = K1 (lanes 0-15) / K3 (lanes 16-31).
  const int m = lane & 15;
  float cmx = cb[m * 3 + 0], cmy = cb[m * 3 + 1], cmz = cb[m * 3 + 2];
  v2f A;
  A.x = (lane < 16) ? (-2.0f * cmx) : (-2.0f * cmz);
  A.y = (lane < 16) ? (-2.0f * cmy) : 1.0f;

  float keptv[16]; int kepti[16]; float rmax[16];
#pragma unroll
  for (int r = 0; r < 16; ++r) { keptv[r] = 3e38f; kepti[r] = 0; rmax[r] = 3e38f; }

  const int col = lane & 15;
  const int hb  = (lane < 16) ? 0 : 2;          // {x,y} vs {z,|x|^2} halves
  const v8f zacc = {};

  for (int n0 = 0; n0 < kN; n0 += 32) {
    // B matrices: one uniform b64 load per tile from the packed float4 ws.
    v2f B0 = *(const v2f*)((const float*)(p4 + (n0 + col)) + hb);
    v2f B1 = *(const v2f*)((const float*)(p4 + (n0 + 16 + col)) + hb);

    v8f c0 = __builtin_amdgcn_wmma_f32_16x16x4_f32(false, A, false, B0,
                                                   (short)0, zacc, false, false);
    v8f c1 = __builtin_amdgcn_wmma_f32_16x16x4_f32(false, A, false, B1,
                                                   (short)0, zacc, false, false);

#pragma unroll
    for (int i = 0; i < 8; ++i) {
      float c0v = c0[i];                        // rows i (lo half) / i+8 (hi half)
      float c1v = c1[i];
      float rm  = (lane < 16) ? rmax[i] : rmax[i + 8];
      unsigned b0m = (unsigned)__ballot(c0v < rm);
      unsigned b1m = (unsigned)__ballot(c1v < rm);
      unsigned mlow  = (b0m & 0xFFFFu) | (b1m << 16);            // row i
      unsigned mhigh = (b0m >> 16) | (b1m & 0xFFFF0000u);        // row i+8
      if (mlow)  INSERT_ROW(i,     mlow,  c0v, c1v);
      if (mhigh) INSERT_ROW(i + 8, mhigh, c0v, c1v);
    }
  }

  // Sort each row's 32 kept entries ascending (value, index) to match
  // jax.lax.top_k's stable order, then gather + subtract center.
#pragma unroll
  for (int r = 0; r < 16; ++r) {
    float v  = keptv[r];
    int   id = kepti[r];
#pragma unroll
    for (int size = 2; size <= 32; size <<= 1) {
#pragma unroll
      for (int stride = size >> 1; stride > 0; stride >>= 1) {
        int partner = lane ^ stride;
        float ov = __shfl(v, partner);
        int   oi = __shfl(id, partner);
        bool up    = ((lane & size) == 0);
        bool lower = ((lane & stride) == 0);
        bool otherSmaller = (ov < v) || (ov == v && oi < id);
        bool keepOther = (up == lower) ? otherSmaller : !otherSmaller;
        if (keepOther) { v = ov; id = oi; }
      }
    }
    float crx = cb[r * 3 + 0], cry = cb[r * 3 + 1], crz = cb[r * 3 + 2];
    float4 p = p4[id];                          // single b128 gather
    size_t o = ((((size_t)b * kG + g0 + r) * kK) + lane) * 3;
    nbr[o + 0] = p.x - crx;
    nbr[o + 1] = p.y - cry;
    nbr[o + 2] = p.z - crz;
  }
}

// ---------------------------------------------------------------------------
extern "C" void kernel_launch(void* const* d_in, const int* in_sizes, int n_in,
                              void* d_out, int out_size, void* d_ws, size_t ws_size,
                              hipStream_t stream) {
  const float* xyz = (const float*)d_in[0];
  float* out = (float*)d_out;
  float* centers = out + (size_t)kB * kG * kK * 3;  // center region of d_out
  float4* x4ws = (float4*)d_ws;                     // kB*kN float4 = 4 MB

  fps_kernel<<<kB, 1024, 0, stream>>>(xyz, x4ws, centers);
  knn_group_kernel<<<kB * (kG / 16), 32, 0, stream>>>(x4ws, centers, out);
}